// GraphConvolution_57690000720131
// MI455X (gfx1250) — compile-verified
//
#include <hip/hip_runtime.h>
#include <hip/hip_bf16.h>

typedef __attribute__((ext_vector_type(2))) float v2f;
typedef __attribute__((ext_vector_type(8))) float v8f;

#define FEAT 128              // F_IN == UNITS == 128
#define TILE_M 128            // rows per workgroup (8 waves x 16)
#define LDS_PAIR_STRIDE 288   // dwords per k-pair row (256 + 32 pad -> disjoint banks per half-wave)

// -------- h = x @ W  (W is [128 x 128] row-major, K-major) --------
// fp32 WMMA 16x16x4: exact fp32 semantics; GEMM is memory-bound anyway.
__global__ __launch_bounds__(256)
void gcn_gemm_kernel(const float* __restrict__ x,
                     const float* __restrict__ W,
                     float* __restrict__ h,
                     int n_rows)
{
    __shared__ float ldsW[64 * LDS_PAIR_STRIDE]; // 72 KB

    // Stage W into LDS as pairs: ldsW[(k>>1)*STRIDE + n*2 + (k&1)]
    // -> each B fragment {W[k][n], W[k+1][n]} is one contiguous float2 (ds_load_b64).
    for (int g = threadIdx.x; g < FEAT * FEAT; g += 256) {
        int k = g >> 7;
        int n = g & 127;
        ldsW[(k >> 1) * LDS_PAIR_STRIDE + n * 2 + (k & 1)] = W[g];
    }
    __syncthreads();

    const int wave  = threadIdx.x >> 5;
    const int lane  = threadIdx.x & 31;
    const int m     = lane & 15;   // M index (A) / N index (B,C)
    const int khalf = lane >> 4;   // which K pair this half-wave holds

    const int rowBase = blockIdx.x * TILE_M + wave * 16;
    int loadRow = rowBase + m;
    if (loadRow >= n_rows) loadRow = n_rows - 1;     // clamp loads; stores guarded
    const float* xrow = x + (size_t)loadRow * FEAT + khalf * 2;

    v8f acc[8];
    #pragma unroll
    for (int i = 0; i < 8; ++i) acc[i] = (v8f){0.f,0.f,0.f,0.f,0.f,0.f,0.f,0.f};

    for (int k0 = 0; k0 < FEAT; k0 += 4) {
        // A 16x4 fp32 fragment: lane(m,khalf) holds x[row][k0+2*khalf + {0,1}]
        v2f a = *(const v2f*)(xrow + k0);
        const float* bp = &ldsW[((k0 >> 1) + khalf) * LDS_PAIR_STRIDE + m * 2];
        #pragma unroll
        for (int nt = 0; nt < 8; ++nt) {
            // B 4x16 fp32 fragment: lane(m,khalf) holds W[k0+2*khalf + {0,1}][nt*16+m]
            v2f b = *(const v2f*)(bp + nt * 32);
            acc[nt] = __builtin_amdgcn_wmma_f32_16x16x4_f32(
                false, a, false, b, (short)0, acc[nt], false, false);
        }
    }

    // C layout: lane L, vgpr r -> (M = r + 8*(L>=16), N = L&15)
    #pragma unroll
    for (int nt = 0; nt < 8; ++nt) {
        #pragma unroll
        for (int r = 0; r < 8; ++r) {
            int row = rowBase + khalf * 8 + r;
            if (row < n_rows)
                h[(size_t)row * FEAT + nt * 16 + m] = acc[nt][r];
        }
    }
}

// -------- out[i, u] = bias[u] --------
__global__ void gcn_init_out(float* __restrict__ out,
                             const float* __restrict__ bias,
                             int total)
{
    int idx = blockIdx.x * blockDim.x + threadIdx.x;
    if (idx < total) out[idx] = bias[idx & (FEAT - 1)];
}

// -------- out[dst[e]] += val[e] * h[src[e]]  (one wave32 per edge) --------
// h (51 MB) and out (51 MB) both fit in the 192 MB L2 -> random traffic stays on-chip.
__global__ __launch_bounds__(256)
void gcn_scatter(const float* __restrict__ h,
                 const int* __restrict__ src,
                 const int* __restrict__ dst,
                 const float* __restrict__ val,
                 float* __restrict__ out,
                 int n_edges)
{
    int wid = (int)((blockIdx.x * (unsigned)blockDim.x + threadIdx.x) >> 5);
    if (wid >= n_edges) return;
    int lane = threadIdx.x & 31;

    int   s = src[wid];
    int   d = dst[wid];
    float v = val[wid];

    const float4 hv = *(const float4*)(h + (size_t)s * FEAT + lane * 4);
    float* op = out + (size_t)d * FEAT + lane * 4;
    atomicAdd(op + 0, hv.x * v);
    atomicAdd(op + 1, hv.y * v);
    atomicAdd(op + 2, hv.z * v);
    atomicAdd(op + 3, hv.w * v);
}

extern "C" void kernel_launch(void* const* d_in, const int* in_sizes, int n_in,
                              void* d_out, int out_size, void* d_ws, size_t ws_size,
                              hipStream_t stream)
{
    const float* x      = (const float*)d_in[0];
    const float* kern   = (const float*)d_in[1];
    const float* bias   = (const float*)d_in[2];
    const int*   esrc   = (const int*)d_in[3];
    const int*   edst   = (const int*)d_in[4];
    const float* eval   = (const float*)d_in[5];

    const int n_nodes = in_sizes[0] / FEAT;
    const int support = in_sizes[1] / (FEAT * FEAT);
    const int n_edges = in_sizes[3] / support;

    float* h = (float*)d_ws; // n_nodes * 128 floats (51.2 MB), reused per support

    // out = bias broadcast (atomics accumulate on top)
    const int total = n_nodes * FEAT;
    gcn_init_out<<<(total + 255) / 256, 256, 0, stream>>>((float*)d_out, bias, total);

    const int gemm_blocks = (n_nodes + TILE_M - 1) / TILE_M;
    const int scat_blocks = (n_edges + 7) / 8; // 8 waves per block, 1 edge per wave
    for (int s = 0; s < support; ++s) {
        gcn_gemm_kernel<<<gemm_blocks, 256, 0, stream>>>(
            x, kern + (size_t)s * FEAT * FEAT, h, n_nodes);
        gcn_scatter<<<scat_blocks, 256, 0, stream>>>(
            h, esrc + (size_t)s * n_edges, edst + (size_t)s * n_edges,
            eval + (size_t)s * n_edges, (float*)d_out, n_edges);
    }
}